// VisualPromptEncoder_25211458028072
// MI455X (gfx1250) — compile-verified
//
#include <hip/hip_runtime.h>

// Problem constants (from the reference)
#define B_    16
#define C_    256
#define H_    100
#define W_    100
#define N_    100
#define RH    40
#define RW    40
#define NCLS  599

typedef __attribute__((ext_vector_type(2))) float v2f;
typedef __attribute__((ext_vector_type(8))) float v8f;

// ---------------------------------------------------------------------------
// Kernel 1: bilinear resize (align_corners=False, no antialias)
// features [B,C,H,W] (row-major) -> resized channel-last [B,RH,RW,C]
// Thread order: (b,c,oy,ox) with ox fastest -> coalesced feature reads.
// Channel-last writes are strided but L2 (192MB) absorbs the 25MB surface.
// ---------------------------------------------------------------------------
__global__ __launch_bounds__(256)
void vpe_resize_kernel(const float* __restrict__ feat, float* __restrict__ resized) {
    int idx = blockIdx.x * blockDim.x + threadIdx.x;
    if (idx >= B_ * C_ * RH * RW) return;
    int ox = idx % RW;  int t = idx / RW;
    int oy = t % RH;    t /= RH;
    int c  = t % C_;    int b = t / C_;

    // half-pixel mapping: src = (dst+0.5)*scale - 0.5, scale = 2.5
    float sx = fmaxf((ox + 0.5f) * ((float)W_ / RW) - 0.5f, 0.0f);
    float sy = fmaxf((oy + 0.5f) * ((float)H_ / RH) - 0.5f, 0.0f);
    int x0 = (int)sx, y0 = (int)sy;
    int x1 = min(x0 + 1, W_ - 1);
    int y1 = min(y0 + 1, H_ - 1);
    float fx = sx - (float)x0, fy = sy - (float)y0;

    const float* p = feat + ((size_t)(b * C_ + c)) * (H_ * W_);
    float v00 = p[y0 * W_ + x0], v01 = p[y0 * W_ + x1];
    float v10 = p[y1 * W_ + x0], v11 = p[y1 * W_ + x1];
    float val = (1.0f - fy) * ((1.0f - fx) * v00 + fx * v01)
              +          fy * ((1.0f - fx) * v10 + fx * v11);

    resized[((size_t)((b * RH + oy) * RW + ox)) * C_ + c] = val;
}

// ---------------------------------------------------------------------------
// Kernel 2: per-(b,box,c) exact box-average pooling over the resized map.
// Boxes on the 40x40 grid are tiny (~1..5 cells/side), so direct summation
// (== integral-image box sum) is ~1..25 coalesced channel-last loads/thread.
// Thread order: (b,n,c) with c fastest -> coalesced reads & writes.
// ---------------------------------------------------------------------------
__global__ __launch_bounds__(256)
void vpe_pool_kernel(const float* __restrict__ resized,
                     const float* __restrict__ boxes,
                     float* __restrict__ pooled,
                     int* __restrict__ validb) {
    int idx = blockIdx.x * blockDim.x + threadIdx.x;
    if (idx >= B_ * N_ * C_) return;
    int c = idx % C_;  int t = idx / C_;
    int n = t % N_;    int b = t / N_;

    const float* bx = boxes + (size_t)(b * N_ + n) * 4;
    const float sW = (float)RW / 1024.0f, sH = (float)RH / 1024.0f;
    // rintf = round-half-to-even under default RNE (matches jnp.round)
    int rx1 = (int)rintf(bx[0] * sW);
    int ry1 = (int)rintf(bx[1] * sH);
    int rx2 = (int)rintf(bx[2] * sW);
    int ry2 = (int)rintf(bx[3] * sH);
    int x1 = max(rx1, 0), y1 = max(ry1, 0);
    int x2 = min(rx2, RW), y2 = min(ry2, RH);
    bool valid = (x1 < x2) && (y1 < y2);
    x1 = min(max(x1, 0), RW);  x2 = min(max(x2, 0), RW);
    y1 = min(max(y1, 0), RH);  y2 = min(max(y2, 0), RH);

    float s = 0.0f;
    if (valid) {
        for (int y = y1; y < y2; ++y)
            for (int x = x1; x < x2; ++x)
                s += resized[((size_t)((b * RH + y) * RW + x)) * C_ + c];
        s /= (float)((y2 - y1) * (x2 - x1));
    }
    pooled[(size_t)(b * N_ + n) * C_ + c] = s;   // zeros for invalid boxes
    if (c == 0) validb[b * N_ + n] = valid ? 1 : 0;
}

// ---------------------------------------------------------------------------
// Kernel 3: scatter-mean as a one-hot GEMM on the matrix pipe.
//   sums[599,256] = onehot[599,100] x pooled[100,256]   (per image)
// One wave per 16x16 (class-tile x channel-tile) output tile, accumulating
// over K=100 boxes in 25 steps of V_WMMA_F32_16X16X4_F32 (f32 is exact for
// the one-hot and preserves reference precision).
// A layout: M = lane&15; K = (lane>>4)*2 + j in VGPR j.
// B layout: N = lane&15; K = (lane>>4)*2 + j in VGPR j.
// D layout: VGPR r -> class row r + 8*(lane>>4), channel col lane&15.
// Per-class valid counts are fetched into D layout with __shfl (ds_bpermute).
// ---------------------------------------------------------------------------
__global__ __launch_bounds__(256)
void vpe_scatter_wmma_kernel(const float* __restrict__ pooled,
                             const int*  __restrict__ gtc,
                             const int*  __restrict__ validb,
                             float* __restrict__ out) {
    const int MT = (NCLS + 15) / 16;   // 38 class tiles (rows 599..607 unused)
    const int NT = C_ / 16;            // 16 channel tiles

    int wave = (int)((blockIdx.x * blockDim.x + threadIdx.x) >> 5);
    int lane = threadIdx.x & 31;
    if (wave >= B_ * MT * NT) return;  // wave-uniform: EXEC stays all-ones

    int nt = wave % NT;  int t = wave / NT;
    int mt = t % MT;     int b = t / MT;
    int m0 = mt * 16, n0 = nt * 16;
    int row  = lane & 15;
    int half = lane >> 5 ? 0 : (lane >> 4);   // lane>>4 in {0,1}
    half = lane >> 4;

    const int* g = gtc    + b * N_;
    const int* v = validb + b * N_;

    // count of valid boxes for class (m0 + row), held in lane "row"
    int mycls = m0 + row;
    float cnt = 0.0f;
    for (int k = 0; k < N_; ++k)
        cnt += ((g[k] == mycls) & v[k]) ? 1.0f : 0.0f;

    const float* P = pooled + (size_t)b * N_ * C_ + n0 + row;

    v8f acc = {};
    #pragma unroll
    for (int kb = 0; kb < N_; kb += 4) {
        int kA = kb + half * 2;
        v2f a, bm;
        // one-hot A (invalid boxes have zeroed pooled rows, so they add 0)
        a.x = (g[kA]     == mycls) ? 1.0f : 0.0f;
        a.y = (g[kA + 1] == mycls) ? 1.0f : 0.0f;
        bm.x = P[(size_t)kA * C_];
        bm.y = P[(size_t)(kA + 1) * C_];
        acc = __builtin_amdgcn_wmma_f32_16x16x4_f32(
                  /*neg_a=*/false, a, /*neg_b=*/false, bm,
                  /*c_mod=*/(short)0, acc, /*reuse_a=*/false, /*reuse_b=*/false);
    }

    #pragma unroll
    for (int r = 0; r < 8; ++r) {
        int cls = m0 + r + 8 * half;
        float cr = __shfl(cnt, r + 8 * half, 32);   // count for this D row
        float val = (cr > 0.0f) ? acc[r] / cr : 0.0f;
        if (cls < NCLS)
            out[((size_t)(b * NCLS + cls)) * C_ + n0 + row] = val;
    }
}

// ---------------------------------------------------------------------------
// Host launcher
// ---------------------------------------------------------------------------
extern "C" void kernel_launch(void* const* d_in, const int* in_sizes, int n_in,
                              void* d_out, int out_size, void* d_ws, size_t ws_size,
                              hipStream_t stream) {
    const float* feat  = (const float*)d_in[0];   // [16,256,100,100] f32
    const float* boxes = (const float*)d_in[1];   // [16,100,4] f32
    const int*   gtc   = (const int*)d_in[2];     // [16,100] i32
    float*       out   = (float*)d_out;           // [16,599,256] f32

    // Workspace carve-up (~27.9 MB total)
    float* resized = (float*)d_ws;                                   // B*RH*RW*C
    float* pooled  = resized + (size_t)B_ * RH * RW * C_;            // B*N*C
    int*   validb  = (int*)(pooled + (size_t)B_ * N_ * C_);          // B*N

    {
        int total = B_ * C_ * RH * RW;
        vpe_resize_kernel<<<(total + 255) / 256, 256, 0, stream>>>(feat, resized);
    }
    {
        int total = B_ * N_ * C_;
        vpe_pool_kernel<<<(total + 255) / 256, 256, 0, stream>>>(resized, boxes, pooled, validb);
    }
    {
        const int MT = (NCLS + 15) / 16, NT = C_ / 16;
        int waves = B_ * MT * NT;                 // 9728 waves
        int threads = waves * 32;                 // 311296, multiple of 256
        vpe_scatter_wmma_kernel<<<(threads + 255) / 256, 256, 0, stream>>>(
            pooled, gtc, validb, out);
    }
}